// MacenkoNormalizer_74672301408557
// MI455X (gfx1250) — compile-verified
//
#include <hip/hip_runtime.h>
#include <hip/hip_bf16.h>
#include <math.h>

// ---------------- problem constants ----------------
#define BCNT   8
#define HW     (1024*1024)
#define NBINS  16384          // phi histogram bins  (64KB LDS)
#define CBINS  8192           // conc histogram bins (2 x 32KB LDS)
#define BETA_V 0.15f
#define ALPHA_PCT 1.0f
#define I0_V   240.0f
#define PI_F   3.14159265358979f
#define LOG_I0 5.4806389233419912f   // log(240)

typedef float v2f __attribute__((ext_vector_type(2)));
typedef float v8f __attribute__((ext_vector_type(8)));

// ---------------- workspace layout (bytes) ----------------
#define OFF_STATS 0            // 8 * 16 f32   (cnt, s0..s2, s00,s01,s02,s11,s12,s22)
#define OFF_EV    512          // 8 * 8 f32    ev[r + 3c], cols = 2 largest eigvecs
#define OFF_PINV  768          // 8 * 8 f32    pinv(he) rows: P[0][0..2], P[1][0..2]
#define OFF_M     1024         // 8 * 12 f32   fused 3x3 output matrix
#define OFF_PHIH  2048         // 8 * 16384 u32
#define OFF_CONCH 526336       // 8 * 2 * 8192 u32
#define WS_BYTES  1050624

// ---------------- WMMA f32 16x16x4 (builtin confirmed on this toolchain) ----------------
#if __has_builtin(__builtin_amdgcn_wmma_f32_16x16x4_f32)
#define HAVE_WMMA_F32X4 1
#else
#define HAVE_WMMA_F32X4 0
#endif

__device__ __forceinline__ v8f wmma_f32_16x16x4(v2f a, v2f b, v8f c) {
#if HAVE_WMMA_F32X4
  // (neg_a, A, neg_b, B, c_mod, C, reuse_a, reuse_b)
  return __builtin_amdgcn_wmma_f32_16x16x4_f32(false, a, false, b, (short)0, c,
                                               false, false);
#else
  v8f d;
  asm volatile("v_wmma_f32_16x16x4_f32 %0, %1, %2, %3"
               : "=&v"(d) : "v"(a), "v"(b), "v"(c));
  return d;
#endif
}

__device__ __forceinline__ float od_of(int x) {
  return LOG_I0 - __logf((float)(x + 1));
}

// ---------------- 0: zero workspace ----------------
__global__ void mk_zero(unsigned* __restrict__ w, int nwords) {
  for (int i = blockIdx.x * blockDim.x + threadIdx.x; i < nwords;
       i += gridDim.x * blockDim.x)
    w[i] = 0u;
}

// ---------------- 1: masked moments ----------------
__global__ void mk_stats(const int* __restrict__ img, float* __restrict__ stats) {
  const int b = blockIdx.y;
  const int* p0 = img + (size_t)b * 3 * HW;
  const int* p1 = p0 + HW;
  const int* p2 = p0 + 2 * HW;
  float a0=0,a1=0,a2=0,a3=0,a4=0,a5=0,a6=0,a7=0,a8=0,a9=0;
  for (int i = blockIdx.x * blockDim.x + threadIdx.x; i < HW;
       i += gridDim.x * blockDim.x) {
    float od0 = od_of(p0[i]);
    float od1 = od_of(p1[i]);
    float od2 = od_of(p2[i]);
    if (od0 >= BETA_V && od1 >= BETA_V && od2 >= BETA_V) {
      a0 += 1.f; a1 += od0; a2 += od1; a3 += od2;
      a4 += od0*od0; a5 += od0*od1; a6 += od0*od2;
      a7 += od1*od1; a8 += od1*od2; a9 += od2*od2;
    }
  }
  __shared__ float s[10];
  if (threadIdx.x < 10) s[threadIdx.x] = 0.f;
  __syncthreads();
  atomicAdd(&s[0], a0); atomicAdd(&s[1], a1); atomicAdd(&s[2], a2);
  atomicAdd(&s[3], a3); atomicAdd(&s[4], a4); atomicAdd(&s[5], a5);
  atomicAdd(&s[6], a6); atomicAdd(&s[7], a7); atomicAdd(&s[8], a8);
  atomicAdd(&s[9], a9);
  __syncthreads();
  if (threadIdx.x < 10)
    atomicAdd(&stats[b * 16 + threadIdx.x], s[threadIdx.x]);
}

// ---------------- 2: 3x3 Jacobi eigensolver (per image) ----------------
__global__ void mk_eigh(const float* __restrict__ stats, float* __restrict__ evout) {
  int b = threadIdx.x;
  if (b >= BCNT) return;
  const float* st = stats + b * 16;
  float n = st[0];
  float inv_n = 1.f / n, inv_nm1 = 1.f / (n - 1.f);
  float m0 = st[1]*inv_n, m1 = st[2]*inv_n, m2 = st[3]*inv_n;
  float A[3][3];
  A[0][0] = (st[4] - n*m0*m0) * inv_nm1;
  A[0][1] = A[1][0] = (st[5] - n*m0*m1) * inv_nm1;
  A[0][2] = A[2][0] = (st[6] - n*m0*m2) * inv_nm1;
  A[1][1] = (st[7] - n*m1*m1) * inv_nm1;
  A[1][2] = A[2][1] = (st[8] - n*m1*m2) * inv_nm1;
  A[2][2] = (st[9] - n*m2*m2) * inv_nm1;
  float V[3][3] = {{1,0,0},{0,1,0},{0,0,1}};
  const int PP[3] = {0,0,1}, QQ[3] = {1,2,2};
  for (int it = 0; it < 24; ++it) {
    int p = PP[it % 3], q = QQ[it % 3];
    float apq = A[p][q];
    if (fabsf(apq) < 1e-20f) continue;
    float theta = 0.5f * (A[q][q] - A[p][p]) / apq;
    float t = copysignf(1.f, theta) / (fabsf(theta) + sqrtf(theta*theta + 1.f));
    float c = 1.f / sqrtf(t*t + 1.f);
    float s = t * c;
    float app = A[p][p], aqq = A[q][q];
    A[p][p] = app - t * apq;
    A[q][q] = aqq + t * apq;
    A[p][q] = A[q][p] = 0.f;
    int k = 3 - p - q;
    float akp = A[k][p], akq = A[k][q];
    A[k][p] = A[p][k] = c*akp - s*akq;
    A[k][q] = A[q][k] = s*akp + c*akq;
    for (int r = 0; r < 3; r++) {
      float vp = V[r][p], vq = V[r][q];
      V[r][p] = c*vp - s*vq;
      V[r][q] = s*vp + c*vq;
    }
  }
  float dd[3] = {A[0][0], A[1][1], A[2][2]};
  int id[3] = {0,1,2};
  if (dd[id[0]] > dd[id[1]]) { int t_=id[0]; id[0]=id[1]; id[1]=t_; }
  if (dd[id[1]] > dd[id[2]]) { int t_=id[1]; id[1]=id[2]; id[2]=t_; }
  if (dd[id[0]] > dd[id[1]]) { int t_=id[0]; id[0]=id[1]; id[1]=t_; }
  float* o = evout + b * 8;
  o[0] = V[0][id[1]]; o[1] = V[1][id[1]]; o[2] = V[2][id[1]];
  o[3] = V[0][id[2]]; o[4] = V[1][id[2]]; o[5] = V[2][id[2]];
}

// ---------------- 3: phi histogram over masked pixels ----------------
__global__ void mk_phihist(const int* __restrict__ img, const float* __restrict__ evin,
                           unsigned* __restrict__ hist) {
  const int b = blockIdx.y;
  __shared__ unsigned h[NBINS];
  for (int i = threadIdx.x; i < NBINS; i += blockDim.x) h[i] = 0u;
  __syncthreads();
  const float* e = evin + b * 8;
  float e00=e[0], e10=e[1], e20=e[2], e01=e[3], e11=e[4], e21=e[5];
  const int* p0 = img + (size_t)b * 3 * HW;
  const int* p1 = p0 + HW;
  const int* p2 = p0 + 2 * HW;
  const float sc = (float)NBINS / (2.f * PI_F);
  for (int i = blockIdx.x * blockDim.x + threadIdx.x; i < HW;
       i += gridDim.x * blockDim.x) {
    float od0 = od_of(p0[i]);
    float od1 = od_of(p1[i]);
    float od2 = od_of(p2[i]);
    if (od0 >= BETA_V && od1 >= BETA_V && od2 >= BETA_V) {
      float t0 = od0*e00 + od1*e10 + od2*e20;
      float t1 = od0*e01 + od1*e11 + od2*e21;
      float phi = atan2f(t1, t0);
      int bin = (int)((phi + PI_F) * sc);
      bin = min(max(bin, 0), NBINS - 1);
      atomicAdd(&h[bin], 1u);
    }
  }
  __syncthreads();
  for (int i = threadIdx.x; i < NBINS; i += blockDim.x)
    if (h[i]) atomicAdd(&hist[(size_t)b * NBINS + i], h[i]);
}

// ---------------- 4: percentile -> HE -> pinv(HE) ----------------
__global__ void mk_he(const float* __restrict__ stats, const float* __restrict__ evin,
                      const unsigned* __restrict__ phih, float* __restrict__ pinv_out) {
  int b = threadIdx.x;
  if (b >= BCNT) return;
  float n = stats[b * 16];
  int kmin = __float2int_rn(0.01f * ALPHA_PCT * (n - 1.f));
  int kmax = __float2int_rn(0.01f * (100.f - ALPHA_PCT) * (n - 1.f));
  const unsigned* h = phih + (size_t)b * NBINS;
  unsigned cum = 0; int bmin = NBINS - 1, bmax = NBINS - 1; bool f1 = false;
  for (int i = 0; i < NBINS; i++) {
    cum += h[i];
    if (!f1 && cum > (unsigned)kmin) { bmin = i; f1 = true; }
    if (cum > (unsigned)kmax) { bmax = i; break; }
  }
  const float bw = 2.f * PI_F / NBINS;
  float minphi = -PI_F + (bmin + 0.5f) * bw;
  float maxphi = -PI_F + (bmax + 0.5f) * bw;
  const float* e = evin + b * 8;
  float sn, cn;
  __sincosf(minphi, &sn, &cn);
  float vn0 = e[0]*cn + e[3]*sn, vn1 = e[1]*cn + e[4]*sn, vn2 = e[2]*cn + e[5]*sn;
  __sincosf(maxphi, &sn, &cn);
  float vx0 = e[0]*cn + e[3]*sn, vx1 = e[1]*cn + e[4]*sn, vx2 = e[2]*cn + e[5]*sn;
  float hA0,hA1,hA2,hB0,hB1,hB2;
  if (vn0 > vx0) { hA0=vn0; hA1=vn1; hA2=vn2; hB0=vx0; hB1=vx1; hB2=vx2; }
  else           { hA0=vx0; hA1=vx1; hA2=vx2; hB0=vn0; hB1=vn1; hB2=vn2; }
  float g00 = hA0*hA0 + hA1*hA1 + hA2*hA2;
  float g01 = hA0*hB0 + hA1*hB1 + hA2*hB2;
  float g11 = hB0*hB0 + hB1*hB1 + hB2*hB2;
  float invdet = 1.f / (g00*g11 - g01*g01);
  float* P = pinv_out + b * 8;
  P[0] = ( g11*hA0 - g01*hB0) * invdet;
  P[1] = ( g11*hA1 - g01*hB1) * invdet;
  P[2] = ( g11*hA2 - g01*hB2) * invdet;
  P[3] = (-g01*hA0 + g00*hB0) * invdet;
  P[4] = (-g01*hA1 + g00*hB1) * invdet;
  P[5] = (-g01*hA2 + g00*hB2) * invdet;
}

// ---------------- 5: concentration histograms (all pixels) ----------------
__global__ void mk_conchist(const int* __restrict__ img, const float* __restrict__ pinv,
                            unsigned* __restrict__ ch) {
  const int b = blockIdx.y;
  __shared__ unsigned h[2 * CBINS];
  for (int i = threadIdx.x; i < 2 * CBINS; i += blockDim.x) h[i] = 0u;
  __syncthreads();
  const float* P = pinv + b * 8;
  float p00=P[0], p01=P[1], p02=P[2], p10=P[3], p11=P[4], p12=P[5];
  const int* q0 = img + (size_t)b * 3 * HW;
  const int* q1 = q0 + HW;
  const int* q2 = q0 + 2 * HW;
  const float sc = (float)CBINS / 32.f;
  for (int i = blockIdx.x * blockDim.x + threadIdx.x; i < HW;
       i += gridDim.x * blockDim.x) {
    float od0 = od_of(q0[i]);
    float od1 = od_of(q1[i]);
    float od2 = od_of(q2[i]);
    float c0 = p00*od0 + p01*od1 + p02*od2;
    float c1 = p10*od0 + p11*od1 + p12*od2;
    int b0 = min(max((int)((c0 + 16.f) * sc), 0), CBINS - 1);
    int b1 = min(max((int)((c1 + 16.f) * sc), 0), CBINS - 1);
    atomicAdd(&h[b0], 1u);
    atomicAdd(&h[CBINS + b1], 1u);
  }
  __syncthreads();
  for (int i = threadIdx.x; i < 2 * CBINS; i += blockDim.x)
    if (h[i]) atomicAdd(&ch[(size_t)b * 2 * CBINS + i], h[i]);
}

// ---------------- 6: max_con -> fused 3x3 matrix M ----------------
__global__ void mk_maxcon(const unsigned* __restrict__ ch, const float* __restrict__ pinv,
                          const float* __restrict__ he_ref, const float* __restrict__ mc_ref,
                          float* __restrict__ Mout) {
  __shared__ float mc[BCNT][2];
  int t = threadIdx.x;
  if (t < 2 * BCNT) {
    int b = t >> 1, s = t & 1;
    const unsigned* h = ch + ((size_t)b * 2 + s) * CBINS;
    const int target = __float2int_rn(0.99f * (float)(HW - 1));  // k99-1 index
    unsigned cum = 0; int bin = CBINS - 1;
    for (int i = 0; i < CBINS; i++) {
      cum += h[i];
      if (cum > (unsigned)target) { bin = i; break; }
    }
    mc[b][s] = -16.f + (bin + 0.5f) * (32.f / (float)CBINS);
  }
  __syncthreads();
  if (t < BCNT) {
    int b = t;
    float s0 = mc_ref[0] / mc[b][0];
    float s1 = mc_ref[1] / mc[b][1];
    const float* P = pinv + b * 8;
    float* M = Mout + b * 12;
    for (int r = 0; r < 3; r++)
      for (int c = 0; c < 3; c++)
        M[r*3 + c] = he_ref[r*2 + 0] * s0 * P[c] + he_ref[r*2 + 1] * s1 * P[3 + c];
  }
}

// ---------------- 7: final color transform via WMMA f32 16x16x4 ----------------
// D = A(16x4) x B(4x16):  A rows 0..2 = M, K = OD channels (K=3 zero);
// B columns = 16 pixels.  B fragments are loaded DIRECTLY in the ISA layout
// (planes are contiguous, so one int index off the image base selects the
// plane per lane) -- no input cross-lane permutes needed.
#define TPAIRS 2   // tile-pairs (64 px) per wave loop iteration

__global__ void mk_final(const int* __restrict__ img, const float* __restrict__ Mw,
                         float* __restrict__ out) {
  const int b = blockIdx.y;
  const int* pb = img + (size_t)b * 3 * HW;
  float* ob = out + (size_t)b * 3 * HW;

  const int lane = threadIdx.x & 31;
  const int wave = threadIdx.x >> 5;
  const bool lo = lane < 16;

  // A fragment (constant over the whole pass):
  // VGPR0: lanes0-15 = A[M=lane][K=0], lanes16-31 = A[M=lane-16][K=2]
  // VGPR1: lanes0-15 = A[M=lane][K=1], lanes16-31 = A[M=lane-16][K=3]=0
  const float* M = Mw + b * 12;
  const int mrow = lane & 15;
  v2f a; a.x = 0.f; a.y = 0.f;
  if (mrow < 3) {
    if (lo) { a.x = M[mrow*3 + 0]; a.y = M[mrow*3 + 1]; }
    else    { a.x = M[mrow*3 + 2]; }
  }
  v8f cz; for (int j = 0; j < 8; j++) cz[j] = 0.f;

  // per-lane plane-selecting indices (relative to pb, planes contiguous)
  const int iA0 = lo ? lane        : (2*HW + lane - 16);  // K0: plane0 | K2: plane2
  const int iA1 = HW + (lane & 15);                       // K1: plane1 (hi lanes dup)
  const int iB0 = lo ? (16 + lane) : (2*HW + lane);
  const int iB1 = HW + 16 + (lane & 15);

  const int gw = blockIdx.x * (blockDim.x >> 5) + wave;
  const int stride = gridDim.x * (blockDim.x >> 5) * (32 * TPAIRS);

  for (int base = gw * (32 * TPAIRS); base < HW; base += stride) {
    v2f bA[TPAIRS], bB[TPAIRS];
#pragma unroll
    for (int t = 0; t < TPAIRS; t++) {
      const int o = base + t * 32;
      bA[t].x = od_of(pb[o + iA0]);
      float yA = od_of(pb[o + iA1]);
      bA[t].y = lo ? yA : 0.f;
      bB[t].x = od_of(pb[o + iB0]);
      float yB = od_of(pb[o + iB1]);
      bB[t].y = lo ? yB : 0.f;
    }
#pragma unroll
    for (int t = 0; t < TPAIRS; t++) {
      const int i = base + t * 32 + lane;
      v8f dA = wmma_f32_16x16x4(a, bA[t], cz);
      v8f dB = wmma_f32_16x16x4(a, bB[t], cz);
      // D row r (VGPR r), lanes 0-15 hold pixel n of each 16-px tile.
      float s0 = __shfl(dB[0], lane & 15);
      float s1 = __shfl(dB[1], lane & 15);
      float s2 = __shfl(dB[2], lane & 15);
      float v0 = lo ? dA[0] : s0;
      float v1 = lo ? dA[1] : s1;
      float v2 = lo ? dA[2] : s2;
      ob[i]        = fminf(I0_V * __expf(-v0), 255.f);
      ob[HW + i]   = fminf(I0_V * __expf(-v1), 255.f);
      ob[2*HW + i] = fminf(I0_V * __expf(-v2), 255.f);
    }
  }
}

// ---------------- launch ----------------
extern "C" void kernel_launch(void* const* d_in, const int* in_sizes, int n_in,
                              void* d_out, int out_size, void* d_ws, size_t ws_size,
                              hipStream_t stream) {
  const int*   img    = (const int*)d_in[0];
  const float* he_ref = (const float*)d_in[1];
  const float* mc_ref = (const float*)d_in[2];
  float* out = (float*)d_out;
  char* ws = (char*)d_ws;

  float*    stats = (float*)(ws + OFF_STATS);
  float*    ev    = (float*)(ws + OFF_EV);
  float*    pinvh = (float*)(ws + OFF_PINV);
  float*    Mmat  = (float*)(ws + OFF_M);
  unsigned* phih  = (unsigned*)(ws + OFF_PHIH);
  unsigned* conch = (unsigned*)(ws + OFF_CONCH);

  mk_zero    <<<256, 256, 0, stream>>>((unsigned*)ws, WS_BYTES / 4);
  mk_stats   <<<dim3(256, BCNT), 256, 0, stream>>>(img, stats);
  mk_eigh    <<<1, 32, 0, stream>>>(stats, ev);
  mk_phihist <<<dim3(128, BCNT), 256, 0, stream>>>(img, ev, phih);
  mk_he      <<<1, 32, 0, stream>>>(stats, ev, phih, pinvh);
  mk_conchist<<<dim3(128, BCNT), 256, 0, stream>>>(img, pinvh, conch);
  mk_maxcon  <<<1, 32, 0, stream>>>(conch, pinvh, he_ref, mc_ref, Mmat);
  mk_final   <<<dim3(256, BCNT), 256, 0, stream>>>(img, Mmat, out);
}